// MySetAggregation_53102975648078
// MI455X (gfx1250) — compile-verified
//
#include <hip/hip_runtime.h>

// ---------------------------------------------------------------------------
// MI455X / gfx1250 stacked SA module (dual-radius ball query + grouping +
// shared MLP w/ batch-stats BN + ReLU + max pool).
//
//  * wave32; ballot/prefix-popcount ordered ball query (one wave per query).
//  * Matmuls: V_WMMA_F32_16X16X4_F32 (f32-exact).  A-frag: lanes 0-15 ->
//    M=lane, K=k0+{0,1}; lanes 16-31 -> M=lane-16, K=k0+{2,3}.  B mirrored.
//    C/D: VGPR r -> (M=r | M=8+r, N=lane&15).
//  * Internal channel order permuted to [feat0..15 | dx dy dz | pad] so the
//    group rows stage into LDS with 4x b128 + 1x b128 vector loads
//    (branch-free; empty groups handled by clamp-index + mask multiply).
//    WMMA fragments then read LDS via aligned ds_load_b64 -> no per-k
//    global-load waits, no EXEC divergence around the WMMAs.
//  * BN needs global per-channel stats between layers -> multi-kernel
//    pipeline; intermediates are recomputed (total math ~1.3 GFLOP, data
//    lives in L2), workspace = index lists + 288 stat floats.
// ---------------------------------------------------------------------------

typedef __attribute__((ext_vector_type(2))) float v2f;
typedef __attribute__((ext_vector_type(8))) float v8f;

#define WPB 8  // waves per 256-thread block (wave32)

static constexpr int   kB    = 2;
static constexpr int   kNper = 16384;
static constexpr int   kMper = 2048;
static constexpr int   kMtot = kB * kMper;   // 4096 query points
static constexpr int   kCin  = 16;
static constexpr int   kKP   = 20;           // layer-0 K padded (19 -> 20)
static constexpr float kEps  = 1e-5f;

__device__ __forceinline__ unsigned uminu(unsigned a, unsigned b) { return a < b ? a : b; }

// ---------------------------------------------------------------------------
// Stage one 16-row group tile into LDS (row-major, kKP floats per row) using
// vectorized loads.  Channel order: [feat 0..15 | dx dy dz | 0].
// Lanes 0-15: features of row (lane); lanes 16-31: xyz-diff of row (lane-16).
// ---------------------------------------------------------------------------
template <int NS>
__device__ __forceinline__ void stage_group_tile(
    float* __restrict__ arow,  // &alds[wid][0]
    int rowbase, const int* __restrict__ gidx, const float* __restrict__ xyz,
    const float* __restrict__ newxyz, const float* __restrict__ feat,
    int lane) {
  const int r   = lane & 15;
  const int row = rowbase + r;
  const int id  = gidx[row];               // flattened [q][s] == row
  const int sid = (id < 0) ? 0 : id;
  const float msk = (id < 0) ? 0.f : 1.f;
  if (lane < 16) {
    const float4* f4 = (const float4*)(feat + sid * kCin);
    float4 a = f4[0], b = f4[1], c = f4[2], d = f4[3];
    a.x *= msk; a.y *= msk; a.z *= msk; a.w *= msk;
    b.x *= msk; b.y *= msk; b.z *= msk; b.w *= msk;
    c.x *= msk; c.y *= msk; c.z *= msk; c.w *= msk;
    d.x *= msk; d.y *= msk; d.z *= msk; d.w *= msk;
    float4* dst = (float4*)(arow + r * kKP);
    dst[0] = a; dst[1] = b; dst[2] = c; dst[3] = d;
  } else {
    const int q = row / NS;
    float4 t;
    t.x = (xyz[sid * 3 + 0] - newxyz[q * 3 + 0]) * msk;
    t.y = (xyz[sid * 3 + 1] - newxyz[q * 3 + 1]) * msk;
    t.z = (xyz[sid * 3 + 2] - newxyz[q * 3 + 2]) * msk;
    t.w = 0.f;
    *(float4*)(arow + r * kKP + 16) = t;
  }
}

// Stage permuted+padded layer-0 weights into LDS: wlds[o*kKP + c'] where
// c'<16 -> W0[o][3+c'] (features), 16..18 -> W0[o][c'-16] (xyz), 19 -> 0.
template <int CO>
__device__ __forceinline__ void stage_w0(float* __restrict__ wlds,
                                         const float* __restrict__ W0,
                                         int tid) {
  for (int i = tid; i < CO * kKP; i += 256) {
    const int o = i / kKP, c = i % kKP;
    float v;
    if (c < 16)       v = W0[o * 19 + 3 + c];
    else if (c < 19)  v = W0[o * 19 + (c - 16)];
    else              v = 0.f;
    wlds[i] = v;
  }
}

// ---------------------------------------------------------------------------
// Kernel 0: copy new_xyz into the first 4096*3 floats of the output.
// ---------------------------------------------------------------------------
__global__ void __launch_bounds__(256) copy_newxyz_kernel(
    const float* __restrict__ src, float* __restrict__ dst) {
  int i = blockIdx.x * 256 + threadIdx.x;
  if (i < kMtot * 3) dst[i] = src[i];
}

// ---------------------------------------------------------------------------
// Kernel 1: dual-radius ball query (one wave / query, ordered via ballot).
// ---------------------------------------------------------------------------
__global__ void __launch_bounds__(256) ballquery_kernel(
    const float* __restrict__ xyz, const float* __restrict__ newxyz,
    int* __restrict__ idx0, int* __restrict__ idx1) {
  __shared__ int sidx[WPB][48];  // [0..15] scale0 slots, [16..47] scale1 slots
  const int lane = threadIdx.x & 31;
  const int wid  = threadIdx.x >> 5;
  const int q    = blockIdx.x * WPB + wid;
  const int base = (q / kMper) * kNper;
  const float nx = newxyz[q * 3 + 0];
  const float ny = newxyz[q * 3 + 1];
  const float nz = newxyz[q * 3 + 2];
  unsigned c0 = 0, c1 = 0;
  const unsigned lt = (1u << lane) - 1u;
  for (int n0 = 0; n0 < kNper; n0 += 32) {
    const int pi   = base + n0 + lane;
    const float dx = xyz[pi * 3 + 0] - nx;
    const float dy = xyz[pi * 3 + 1] - ny;
    const float dz = xyz[pi * 3 + 2] - nz;
    const float d2 = dx * dx + dy * dy + dz * dz;
    const bool in0 = d2 < 0.01f;  // r=0.1
    const bool in1 = d2 < 0.04f;  // r=0.2
    const unsigned m0 = __builtin_amdgcn_ballot_w32(in0);
    const unsigned m1 = __builtin_amdgcn_ballot_w32(in1);
    if (in0) { unsigned s = c0 + (unsigned)__popc(m0 & lt); if (s < 16u) sidx[wid][s] = pi; }
    if (in1) { unsigned s = c1 + (unsigned)__popc(m1 & lt); if (s < 32u) sidx[wid][16 + s] = pi; }
    c0 += (unsigned)__popc(m0);
    c1 += (unsigned)__popc(m1);
    if (c0 >= 16u && c1 >= 32u) break;  // wave-uniform
  }
  __syncthreads();
  const int f0 = (c0 > 0u) ? sidx[wid][0]  : -1;
  const int f1 = (c1 > 0u) ? sidx[wid][16] : -1;
  if (lane < 16) {
    int v = (c0 == 0u) ? -1 : ((unsigned)lane < uminu(c0, 16u) ? sidx[wid][lane] : f0);
    idx0[q * 16 + lane] = v;
  }
  {
    int v = (c1 == 0u) ? -1 : ((unsigned)lane < uminu(c1, 32u) ? sidx[wid][16 + lane] : f1);
    idx1[q * 32 + lane] = v;
  }
}

// ---------------------------------------------------------------------------
// Kernel 2: layer-0 pre-BN statistics.  Wave: stage 16-row A tile in LDS,
// loop over CO/16 col tiles, 5 WMMA k-steps each; channel stats via LDS
// ds_add_f32 then one global atomic per channel per block.
// ---------------------------------------------------------------------------
template <int NS, int CO>
__global__ void __launch_bounds__(256) layer0_stats_kernel(
    const int* __restrict__ gidx, const float* __restrict__ xyz,
    const float* __restrict__ newxyz, const float* __restrict__ feat,
    const float* __restrict__ W0, float* __restrict__ gsum,
    float* __restrict__ gsq) {
  __shared__ float alds[WPB][16 * kKP];
  __shared__ float wlds[CO * kKP];
  __shared__ float ssum[CO], ssq[CO];
  const int tid = threadIdx.x;
  if (tid < CO) { ssum[tid] = 0.f; ssq[tid] = 0.f; }
  stage_w0<CO>(wlds, W0, tid);
  const int lane = tid & 31, wid = tid >> 5;
  const int rt = blockIdx.x * WPB + wid;
  stage_group_tile<NS>(&alds[wid][0], rt * 16, gidx, xyz, newxyz, feat, lane);
  __syncthreads();
  const int m  = lane & 15;
  const int kofs = (lane >= 16) ? 2 : 0;
#pragma unroll
  for (int ct = 0; ct < CO / 16; ++ct) {
    const int n = ct * 16 + m;
    v8f acc = {};
#pragma unroll
    for (int k0 = 0; k0 < kKP; k0 += 4) {
      const int ka = k0 + kofs;
      v2f a = *(const v2f*)&alds[wid][m * kKP + ka];
      v2f b = *(const v2f*)&wlds[n * kKP + ka];
      acc = __builtin_amdgcn_wmma_f32_16x16x4_f32(false, a, false, b, (short)0,
                                                  acc, false, false);
    }
    float ps = 0.f, ps2 = 0.f;
#pragma unroll
    for (int r = 0; r < 8; ++r) { float v = acc[r]; ps += v; ps2 += v * v; }
    atomicAdd(&ssum[n], ps);   // all 8 acc elems of a lane share channel n
    atomicAdd(&ssq[n], ps2);
  }
  __syncthreads();
  if (tid < CO) { atomicAdd(&gsum[tid], ssum[tid]); atomicAdd(&gsq[tid], ssq[tid]); }
}

// ---------------------------------------------------------------------------
// Kernel 3: layer-1 pre-BN statistics.  Recompute Y tile (A from LDS stage),
// BN0+ReLU, stage normalized tile, run Z = Yn @ W1^T, accumulate stats.
// ---------------------------------------------------------------------------
template <int NS, int C1, int C2>
__global__ void __launch_bounds__(256) layer1_stats_kernel(
    const int* __restrict__ gidx, const float* __restrict__ xyz,
    const float* __restrict__ newxyz, const float* __restrict__ feat,
    const float* __restrict__ W0, const float* __restrict__ gamma0,
    const float* __restrict__ beta0, const float* __restrict__ W1,
    const float* __restrict__ sum0, const float* __restrict__ sq0,
    float* __restrict__ gsum, float* __restrict__ gsq) {
  constexpr int ROWS = kMtot * NS;
  __shared__ float alds[WPB][16 * kKP];
  __shared__ float wlds0[C1 * kKP];
  __shared__ float wlds1[C2 * C1];
  __shared__ float ylds[WPB][16 * C1];
  __shared__ float sc0[C1], sh0[C1];
  __shared__ float ssum[C2], ssq[C2];
  const int tid = threadIdx.x;
  if (tid < C2) { ssum[tid] = 0.f; ssq[tid] = 0.f; }
  if (tid < C1) {
    const float invn = 1.0f / (float)ROWS;
    float mean = sum0[tid] * invn;
    float var  = sq0[tid] * invn - mean * mean;
    float sfac = gamma0[tid] * rsqrtf(var + kEps);
    sc0[tid] = sfac;
    sh0[tid] = beta0[tid] - mean * sfac;
  }
  stage_w0<C1>(wlds0, W0, tid);
  for (int i = tid; i < C2 * C1; i += 256) wlds1[i] = W1[i];
  const int lane = tid & 31, wid = tid >> 5;
  const int rt = blockIdx.x * WPB + wid;
  stage_group_tile<NS>(&alds[wid][0], rt * 16, gidx, xyz, newxyz, feat, lane);
  __syncthreads();
  const int m  = lane & 15;
  const bool kh = lane >= 16;
  const int kofs = kh ? 2 : 0;
  // --- Y row-tile, normalize+ReLU, stage in LDS ---
#pragma unroll
  for (int ct = 0; ct < C1 / 16; ++ct) {
    const int n = ct * 16 + m;
    v8f acc = {};
#pragma unroll
    for (int k0 = 0; k0 < kKP; k0 += 4) {
      const int ka = k0 + kofs;
      v2f a = *(const v2f*)&alds[wid][m * kKP + ka];
      v2f b = *(const v2f*)&wlds0[n * kKP + ka];
      acc = __builtin_amdgcn_wmma_f32_16x16x4_f32(false, a, false, b, (short)0,
                                                  acc, false, false);
    }
#pragma unroll
    for (int r = 0; r < 8; ++r) {
      const int rr = kh ? (8 + r) : r;
      ylds[wid][rr * C1 + n] = fmaxf(acc[r] * sc0[n] + sh0[n], 0.f);
    }
  }
  __syncthreads();
  // --- Z = Yn @ W1^T, accumulate channel stats ---
#pragma unroll
  for (int ct = 0; ct < C2 / 16; ++ct) {
    const int n = ct * 16 + m;
    v8f acc = {};
#pragma unroll
    for (int k0 = 0; k0 < C1; k0 += 4) {
      const int ka = k0 + kofs;
      v2f a = *(const v2f*)&ylds[wid][m * C1 + ka];
      v2f b = *(const v2f*)&wlds1[n * C1 + ka];
      acc = __builtin_amdgcn_wmma_f32_16x16x4_f32(false, a, false, b, (short)0,
                                                  acc, false, false);
    }
    float ps = 0.f, ps2 = 0.f;
#pragma unroll
    for (int r = 0; r < 8; ++r) { float v = acc[r]; ps += v; ps2 += v * v; }
    atomicAdd(&ssum[n], ps);
    atomicAdd(&ssq[n], ps2);
  }
  __syncthreads();
  if (tid < C2) { atomicAdd(&gsum[tid], ssum[tid]); atomicAdd(&gsq[tid], ssq[tid]); }
}

// ---------------------------------------------------------------------------
// Kernel 4: final pass.  One wave per query; NS/16 row tiles: stage A, Y tile,
// BN0+ReLU, Z tiles, BN1+ReLU, running max in registers; half-wave combine
// via ds_bpermute; write pooled (4096 x C2) slice of the output.
// ---------------------------------------------------------------------------
template <int NS, int C1, int C2, int COFF>
__global__ void __launch_bounds__(256) pool_kernel(
    const int* __restrict__ gidx, const float* __restrict__ xyz,
    const float* __restrict__ newxyz, const float* __restrict__ feat,
    const float* __restrict__ W0, const float* __restrict__ gamma0,
    const float* __restrict__ beta0, const float* __restrict__ W1,
    const float* __restrict__ gamma1, const float* __restrict__ beta1,
    const float* __restrict__ sum0, const float* __restrict__ sq0,
    const float* __restrict__ sum1, const float* __restrict__ sq1,
    float* __restrict__ outfeat) {
  constexpr int ROWS = kMtot * NS;
  __shared__ float alds[WPB][16 * kKP];
  __shared__ float wlds0[C1 * kKP];
  __shared__ float wlds1[C2 * C1];
  __shared__ float ylds[WPB][16 * C1];
  __shared__ float sc0[C1], sh0[C1], sc1[C2], sh1[C2];
  const int tid = threadIdx.x;
  const float invn = 1.0f / (float)ROWS;
  if (tid < C1) {
    float mean = sum0[tid] * invn;
    float var  = sq0[tid] * invn - mean * mean;
    float sfac = gamma0[tid] * rsqrtf(var + kEps);
    sc0[tid] = sfac;
    sh0[tid] = beta0[tid] - mean * sfac;
  }
  if (tid < C2) {
    float mean = sum1[tid] * invn;
    float var  = sq1[tid] * invn - mean * mean;
    float sfac = gamma1[tid] * rsqrtf(var + kEps);
    sc1[tid] = sfac;
    sh1[tid] = beta1[tid] - mean * sfac;
  }
  stage_w0<C1>(wlds0, W0, tid);
  for (int i = tid; i < C2 * C1; i += 256) wlds1[i] = W1[i];
  const int lane = tid & 31, wid = tid >> 5;
  const int q  = blockIdx.x * WPB + wid;
  const int m  = lane & 15;
  const bool kh = lane >= 16;
  const int kofs = kh ? 2 : 0;
  float vmax[C2 / 16][8];
#pragma unroll
  for (int ct = 0; ct < C2 / 16; ++ct)
#pragma unroll
    for (int r = 0; r < 8; ++r) vmax[ct][r] = 0.f;  // post-ReLU values >= 0
  __syncthreads();

#pragma unroll
  for (int rti = 0; rti < NS / 16; ++rti) {
    const int rowbase = (q * (NS / 16) + rti) * 16;
    stage_group_tile<NS>(&alds[wid][0], rowbase, gidx, xyz, newxyz, feat, lane);
    __syncthreads();
    // Y tile -> BN0+ReLU -> LDS
#pragma unroll
    for (int ct = 0; ct < C1 / 16; ++ct) {
      const int n = ct * 16 + m;
      v8f acc = {};
#pragma unroll
      for (int k0 = 0; k0 < kKP; k0 += 4) {
        const int ka = k0 + kofs;
        v2f a = *(const v2f*)&alds[wid][m * kKP + ka];
        v2f b = *(const v2f*)&wlds0[n * kKP + ka];
        acc = __builtin_amdgcn_wmma_f32_16x16x4_f32(false, a, false, b, (short)0,
                                                    acc, false, false);
      }
#pragma unroll
      for (int r = 0; r < 8; ++r) {
        const int rr = kh ? (8 + r) : r;
        ylds[wid][rr * C1 + n] = fmaxf(acc[r] * sc0[n] + sh0[n], 0.f);
      }
    }
    __syncthreads();
    // Z tiles -> BN1 + ReLU -> running max
#pragma unroll
    for (int ct = 0; ct < C2 / 16; ++ct) {
      const int n = ct * 16 + m;
      v8f acc = {};
#pragma unroll
      for (int k0 = 0; k0 < C1; k0 += 4) {
        const int ka = k0 + kofs;
        v2f a = *(const v2f*)&ylds[wid][m * C1 + ka];
        v2f b = *(const v2f*)&wlds1[n * C1 + ka];
        acc = __builtin_amdgcn_wmma_f32_16x16x4_f32(false, a, false, b, (short)0,
                                                    acc, false, false);
      }
#pragma unroll
      for (int r = 0; r < 8; ++r) {
        float v = fmaxf(acc[r] * sc1[n] + sh1[n], 0.f);
        vmax[ct][r] = fmaxf(vmax[ct][r], v);
      }
    }
    __syncthreads();
  }
  // reduce over 8 local rows, then combine lane <-> lane^16 (M 0-7 vs 8-15)
#pragma unroll
  for (int ct = 0; ct < C2 / 16; ++ct) {
    float mm = vmax[ct][0];
#pragma unroll
    for (int r = 1; r < 8; ++r) mm = fmaxf(mm, vmax[ct][r]);
    int oi = __builtin_amdgcn_ds_bpermute(((lane ^ 16) << 2), __float_as_int(mm));
    float tot = fmaxf(mm, __int_as_float(oi));
    if (lane < 16) outfeat[q * 96 + COFF + ct * 16 + lane] = tot;
  }
}

// ---------------------------------------------------------------------------
extern "C" void kernel_launch(void* const* d_in, const int* in_sizes, int n_in,
                              void* d_out, int out_size, void* d_ws,
                              size_t ws_size, hipStream_t stream) {
  (void)in_sizes; (void)n_in; (void)out_size; (void)ws_size;
  const float* xyz    = (const float*)d_in[0];
  const float* newxyz = (const float*)d_in[2];
  const float* feat   = (const float*)d_in[4];
  const float* w0_0 = (const float*)d_in[5];
  const float* g0_0 = (const float*)d_in[6];
  const float* b0_0 = (const float*)d_in[7];
  const float* w0_1 = (const float*)d_in[8];
  const float* g0_1 = (const float*)d_in[9];
  const float* b0_1 = (const float*)d_in[10];
  const float* w1_0 = (const float*)d_in[11];
  const float* g1_0 = (const float*)d_in[12];
  const float* b1_0 = (const float*)d_in[13];
  const float* w1_1 = (const float*)d_in[14];
  const float* g1_1 = (const float*)d_in[15];
  const float* b1_1 = (const float*)d_in[16];

  float* out     = (float*)d_out;
  float* outfeat = out + kMtot * 3;

  int*   idx0 = (int*)d_ws;                 // 4096*16 ints
  int*   idx1 = idx0 + kMtot * 16;          // 4096*32 ints
  float* st   = (float*)(idx1 + kMtot * 32);
  float* s0l0_sum = st;        float* s0l0_sq = st + 16;
  float* s0l1_sum = st + 32;   float* s0l1_sq = st + 64;
  float* s1l0_sum = st + 96;   float* s1l0_sq = st + 128;
  float* s1l1_sum = st + 160;  float* s1l1_sq = st + 224;  // 288 floats total

  hipMemsetAsync(st, 0, 288 * sizeof(float), stream);

  copy_newxyz_kernel<<<(kMtot * 3 + 255) / 256, 256, 0, stream>>>(newxyz, out);
  ballquery_kernel<<<kMtot / WPB, 256, 0, stream>>>(xyz, newxyz, idx0, idx1);

  // scale 0: NS=16, 19->16->32   scale 1: NS=32, 19->32->64
  layer0_stats_kernel<16, 16><<<4096 / WPB, 256, 0, stream>>>(
      idx0, xyz, newxyz, feat, w0_0, s0l0_sum, s0l0_sq);
  layer0_stats_kernel<32, 32><<<8192 / WPB, 256, 0, stream>>>(
      idx1, xyz, newxyz, feat, w1_0, s1l0_sum, s1l0_sq);

  layer1_stats_kernel<16, 16, 32><<<4096 / WPB, 256, 0, stream>>>(
      idx0, xyz, newxyz, feat, w0_0, g0_0, b0_0, w0_1,
      s0l0_sum, s0l0_sq, s0l1_sum, s0l1_sq);
  layer1_stats_kernel<32, 32, 64><<<8192 / WPB, 256, 0, stream>>>(
      idx1, xyz, newxyz, feat, w1_0, g1_0, b1_0, w1_1,
      s1l0_sum, s1l0_sq, s1l1_sum, s1l1_sq);

  pool_kernel<16, 16, 32, 0><<<kMtot / WPB, 256, 0, stream>>>(
      idx0, xyz, newxyz, feat, w0_0, g0_0, b0_0, w0_1, g0_1, b0_1,
      s0l0_sum, s0l0_sq, s0l1_sum, s0l1_sq, outfeat);
  pool_kernel<32, 32, 64, 32><<<kMtot / WPB, 256, 0, stream>>>(
      idx1, xyz, newxyz, feat, w1_0, g1_0, b1_0, w1_1, g1_1, b1_1,
      s1l0_sum, s1l0_sq, s1l1_sum, s1l1_sq, outfeat);
}